// NetVLAD_52621939310770
// MI455X (gfx1250) — compile-verified
//
#include <hip/hip_runtime.h>
#include <math.h>

#define NN 32
#define DD 128
#define PP 4096
#define KK 64
#define PT 256                     // pixels per tile
#define TILES_PER_IMG (PP / PT)    // 16
#define XS 257                     // xn LDS row stride (odd -> conflict-free columns)
#define WSRD 129                   // W LDS row stride
#define SS 257                     // scores LDS row stride
#define EPSN 1e-12f

typedef float v2f __attribute__((ext_vector_type(2)));
typedef float v8f __attribute__((ext_vector_type(8)));

__global__ void zero_ws_kernel(float* __restrict__ p, int n) {
    int i = blockIdx.x * blockDim.x + threadIdx.x;
    if (i < n) p[i] = 0.0f;
}

__global__ __launch_bounds__(256) void netvlad_phase1(
    const float* __restrict__ x,     // [N, D, P]
    const float* __restrict__ w,     // [K, D]
    float* __restrict__ pv,          // [N, K, D] partial vlad (atomic accum)
    float* __restrict__ asum_g)      // [N, K]    partial assignment sums
{
    extern __shared__ float smem[];
    float* Xl  = smem;               // [DD][XS]  normalized x tile
    float* Wl  = Xl + DD * XS;       // [KK][WSRD]
    float* Sl  = Wl + KK * WSRD;     // [KK][SS]  scores -> softmax a
    float* asp = Sl + KK * SS;       // [256] partial cluster sums

    const int t    = threadIdx.x;
    const int img  = blockIdx.x / TILES_PER_IMG;
    const int p0   = (blockIdx.x % TILES_PER_IMG) * PT;
    const int lane = t & 31;
    const int wv   = t >> 5;
    const int fm   = lane & 15;      // M/N index within 16x16 tile
    const int fh   = lane >> 4;      // K half selector

    // ---- load W into LDS (8192 floats / 256 threads) ----
    for (int j = 0; j < (KK * DD) / 256; ++j) {
        int idx = j * 256 + t;
        Wl[(idx >> 7) * WSRD + (idx & 127)] = w[idx];
    }

    // ---- load x tile + per-pixel L2 normalization (thread t owns pixel p0+t) ----
    const float* xb = x + (size_t)img * DD * PP + p0;
    float ss = 0.0f;
    for (int d = 0; d < DD; ++d) {
        float v = xb[(size_t)d * PP + t];
        ss += v * v;
        Xl[d * XS + t] = v;
    }
    float invn = 1.0f / fmaxf(sqrtf(ss), EPSN);
    for (int d = 0; d < DD; ++d) Xl[d * XS + t] *= invn;
    __syncthreads();

    // ---- GEMM1: Sl(64x256) = Wl(64x128) @ Xl(128x256), f32 WMMA 16x16x4 ----
    // Each wave owns one 16-row M-tile (mt = wv>>1) and 8 N-tiles.
    // A fragment loaded once per k-step, reused across all 8 N-tiles.
    {
        const int mt = wv >> 1;
        const int nb = (wv & 1) * 8;         // first of 8 N-tiles
        const int arow = (mt * 16 + fm) * WSRD;
        v8f acc[8];
        #pragma unroll
        for (int i = 0; i < 8; ++i)
            acc[i] = (v8f){0.f, 0.f, 0.f, 0.f, 0.f, 0.f, 0.f, 0.f};

        for (int kb = 0; kb < DD; kb += 4) {
            const int kc = kb + 2 * fh;
            v2f A;
            A.x = Wl[arow + kc];
            A.y = Wl[arow + kc + 1];
            const float* b0 = Xl + kc * XS + nb * 16 + fm;
            const float* b1 = Xl + (kc + 1) * XS + nb * 16 + fm;
            #pragma unroll
            for (int i = 0; i < 8; ++i) {
                v2f B;
                B.x = b0[i * 16];
                B.y = b1[i * 16];
                acc[i] = __builtin_amdgcn_wmma_f32_16x16x4_f32(
                    false, A, false, B, (short)0, acc[i], false, false);
            }
        }
        #pragma unroll
        for (int i = 0; i < 8; ++i)
            #pragma unroll
            for (int r = 0; r < 8; ++r)
                Sl[(mt * 16 + r + 8 * fh) * SS + (nb + i) * 16 + fm] = acc[i][r];
    }
    __syncthreads();

    // ---- softmax over clusters for pixel t (in-place in Sl) ----
    {
        float* col = Sl + t;
        float m = -3.402823466e38f;
        for (int k = 0; k < KK; ++k) m = fmaxf(m, col[k * SS]);
        float s = 0.0f;
        for (int k = 0; k < KK; ++k) {
            float e = __expf(col[k * SS] - m);
            col[k * SS] = e;
            s += e;
        }
        float r = 1.0f / s;
        for (int k = 0; k < KK; ++k) col[k * SS] *= r;
    }
    __syncthreads();

    // ---- per-cluster assignment sums: asum[k] = sum_p a[k][p] ----
    {
        int q = t >> 6, k = t & 63;
        float s = 0.0f;
        const float* row = Sl + k * SS + q * 64;
        for (int j = 0; j < 64; ++j) s += row[j];
        asp[q * 64 + k] = s;
    }
    __syncthreads();
    if (t < KK) {
        float s = asp[t] + asp[64 + t] + asp[128 + t] + asp[192 + t];
        __hip_atomic_fetch_add(&asum_g[img * KK + t], s,
                               __ATOMIC_RELAXED, __HIP_MEMORY_SCOPE_AGENT);
    }

    // ---- GEMM2: pv(64x128) += a(64x256) @ xn^T(256x128) ----
    // Each wave owns one M-tile (mt = wv>>1) and 4 N-tiles; A reused per k-step.
    {
        const int mt = wv >> 1;
        const int nb = (wv & 1) * 4;         // first of 4 N-tiles
        const int arow = (mt * 16 + fm) * SS;
        v8f acc[4];
        #pragma unroll
        for (int i = 0; i < 4; ++i)
            acc[i] = (v8f){0.f, 0.f, 0.f, 0.f, 0.f, 0.f, 0.f, 0.f};

        for (int kb = 0; kb < PT; kb += 4) {
            const int kc = kb + 2 * fh;
            v2f A;
            A.x = Sl[arow + kc];             // a[cluster][pixel]
            A.y = Sl[arow + kc + 1];
            #pragma unroll
            for (int i = 0; i < 4; ++i) {
                const float* brow = Xl + ((nb + i) * 16 + fm) * XS;  // B(p,d)=xn[d][p]
                v2f B;
                B.x = brow[kc];
                B.y = brow[kc + 1];
                acc[i] = __builtin_amdgcn_wmma_f32_16x16x4_f32(
                    false, A, false, B, (short)0, acc[i], false, false);
            }
        }
        #pragma unroll
        for (int i = 0; i < 4; ++i)
            #pragma unroll
            for (int r = 0; r < 8; ++r) {
                int kcl = mt * 16 + r + 8 * fh;
                int dd  = (nb + i) * 16 + fm;
                __hip_atomic_fetch_add(&pv[((size_t)img * KK + kcl) * DD + dd],
                                       acc[i][r], __ATOMIC_RELAXED,
                                       __HIP_MEMORY_SCOPE_AGENT);
            }
    }
}

__global__ __launch_bounds__(256) void netvlad_phase2(
    const float* __restrict__ pv,
    const float* __restrict__ asum_g,
    const float* __restrict__ cent,   // [K, D]
    float* __restrict__ out)          // [N, K*D]
{
    __shared__ float ob[KK * DD];
    __shared__ float asl[KK];
    __shared__ float rss[KK];
    __shared__ float gtot;
    const int t = threadIdx.x;
    const int n = blockIdx.x;
    const int lane = t & 31;
    const int wv = t >> 5;

    if (t < KK) asl[t] = asum_g[n * KK + t];
    __syncthreads();

    // each wave handles 8 cluster rows; lane owns 4 channels
    for (int j = 0; j < 8; ++j) {
        int k  = wv * 8 + j;
        int d0 = lane * 4;
        const float* pr = pv + ((size_t)n * KK + k) * DD + d0;
        const float* cr = cent + k * DD + d0;
        float a = asl[k];
        float v0 = pr[0] - cr[0] * a;
        float v1 = pr[1] - cr[1] * a;
        float v2 = pr[2] - cr[2] * a;
        float v3 = pr[3] - cr[3] * a;
        float ss = v0 * v0 + v1 * v1 + v2 * v2 + v3 * v3;
        for (int m = 16; m > 0; m >>= 1) ss += __shfl_xor(ss, m, 32);
        float inv = 1.0f / fmaxf(sqrtf(ss), EPSN);
        ob[k * DD + d0 + 0] = v0 * inv;
        ob[k * DD + d0 + 1] = v1 * inv;
        ob[k * DD + d0 + 2] = v2 * inv;
        ob[k * DD + d0 + 3] = v3 * inv;
        if (lane == 0) rss[k] = ss * inv * inv;   // ||row||^2 after intra-norm
    }
    __syncthreads();
    if (t == 0) {
        float g = 0.0f;
        for (int i = 0; i < KK; ++i) g += rss[i];
        gtot = g;
    }
    __syncthreads();
    float ginv = 1.0f / fmaxf(sqrtf(gtot), EPSN);
    for (int j = 0; j < (KK * DD) / 256; ++j) {
        int idx = j * 256 + t;
        out[(size_t)n * KK * DD + idx] = ob[idx] * ginv;
    }
}

extern "C" void kernel_launch(void* const* d_in, const int* in_sizes, int n_in,
                              void* d_out, int out_size, void* d_ws, size_t ws_size,
                              hipStream_t stream) {
    const float* x = (const float*)d_in[0];   // [32,128,64,64]
    const float* w = (const float*)d_in[1];   // [64,128]
    const float* c = (const float*)d_in[2];   // [64,128]
    float* out = (float*)d_out;

    float* pv = (float*)d_ws;                         // [32,64,128]
    float* as = pv + (size_t)NN * KK * DD;            // [32,64]
    const int zn = NN * KK * DD + NN * KK;

    zero_ws_kernel<<<(zn + 255) / 256, 256, 0, stream>>>(pv, zn);

    size_t shmem = (size_t)(DD * XS + KK * WSRD + KK * SS + 256) * sizeof(float);
    netvlad_phase1<<<NN * TILES_PER_IMG, 256, shmem, stream>>>(x, w, pv, as);
    netvlad_phase2<<<NN, 256, 0, stream>>>(pv, as, c, out);
}